// MaskCNN_29764123361495
// MI455X (gfx1250) — compile-verified
//
#include <hip/hip_runtime.h>
#include <hip/hip_bf16.h>

typedef float v2f __attribute__((ext_vector_type(2)));
typedef float v8f __attribute__((ext_vector_type(8)));

#define BN_EPS 1e-5f

// ---------------------------------------------------------------------------
// Layer 1: Conv2d(1,32,(41,11),s=(2,2),p=(20,5)) + mask/BN/hardtanh fused.
// Implicit GEMM, M=32 (o), N=spatial (oh fixed per block, 128 ow per block),
// K = kh x kw(11 padded to 12). Each wave: 32x16 tile via two
// v_wmma_f32_16x16x4_f32 accumulator chains; 3 K-chunks of 4 per kh row.
// The ih-valid region is a contiguous kh interval -> folded into the loop
// bounds (no branch in the body, EXEC all-ones, no accumulator phi copies).
// Column guard is branch-free: wrap-masked load + cndmask.
// ---------------------------------------------------------------------------
__global__ __launch_bounds__(256) void maskcnn_conv1(
    const float* __restrict__ in,    // [32,1,161,1024]
    const int*   __restrict__ seq,   // [32]
    const float* __restrict__ w1,    // [32,1,41,11]
    const float* __restrict__ b1,
    const float* __restrict__ g1, const float* __restrict__ be1,
    const float* __restrict__ m1, const float* __restrict__ v1,
    float* __restrict__ out)         // [32,32,81,512] (workspace)
{
    __shared__ float Wl[32 * 41 * 12];   // kw padded 11 -> 12 (pad weight = 0)
    __shared__ float sScale[32];
    __shared__ float sShift[32];

    const int tid   = threadIdx.x;
    const int b     = blockIdx.z;
    const int oh    = blockIdx.y;
    const int owblk = blockIdx.x * 128;
    const int len1  = ((seq[b] - 1) >> 1) + 1;

    // Fully-masked tile: emit zeros, skip all compute (uniform branch).
    if (owblk >= len1) {
        for (int idx = tid; idx < 32 * 128; idx += 256) {
            int o  = idx >> 7;
            int ow = owblk + (idx & 127);
            out[(((size_t)b * 32 + o) * 81 + oh) * 512 + ow] = 0.0f;
        }
        return;
    }

    for (int idx = tid; idx < 32 * 492; idx += 256) {
        int o = idx / 492, r = idx - o * 492;
        int kh = r / 12, kw = r - kh * 12;
        Wl[idx] = (kw < 11) ? w1[o * 451 + kh * 11 + kw] : 0.0f;
    }
    if (tid < 32) {
        float inv = g1[tid] * __frsqrt_rn(v1[tid] + BN_EPS);
        sScale[tid] = inv;
        sShift[tid] = b1[tid] * inv + be1[tid] - m1[tid] * inv;
    }
    __syncthreads();

    const int wave   = tid >> 5;
    const int lane   = tid & 31;
    const int half16 = lane >> 4;
    const int m      = lane & 15;
    const int ow_n   = owblk + wave * 16 + m;    // this lane's N column
    const int iwbase = 2 * ow_n - 5;

    // ih = 2*oh - 20 + kh in [0,161)  <=>  kh in [20-2*oh, 181-2*oh)
    const int kh_lo = max(0, 20 - 2 * oh);
    const int kh_hi = min(41, 181 - 2 * oh);

    const float* __restrict__ inb = in + (size_t)b * 161 * 1024;
    const float* __restrict__ wl0 = Wl + m * 492;
    const float* __restrict__ wl1 = Wl + (m + 16) * 492;

    v8f c0 = {};
    v8f c1 = {};

    for (int kh = kh_lo; kh < kh_hi; ++kh) {
        const int ih = 2 * oh - 20 + kh;         // always in range
        const float* __restrict__ row = inb + ih * 1024;
#pragma unroll
        for (int c = 0; c < 3; ++c) {
            const int kb = c * 4 + half16 * 2;   // lane's kw pair base
            v2f bf;
#pragma unroll
            for (int j = 0; j < 2; ++j) {
                int iw = iwbase + kb + j;
                float v = row[iw & 1023];         // wrap-safe address
                bf[j] = ((unsigned)iw < 1024u) ? v : 0.0f;
            }
            v2f a0, a1;
            a0[0] = wl0[kh * 12 + kb]; a0[1] = wl0[kh * 12 + kb + 1];
            a1[0] = wl1[kh * 12 + kb]; a1[1] = wl1[kh * 12 + kb + 1];

            c0 = __builtin_amdgcn_wmma_f32_16x16x4_f32(false, a0, false, bf,
                                                       (short)0, c0, false, false);
            c1 = __builtin_amdgcn_wmma_f32_16x16x4_f32(false, a1, false, bf,
                                                       (short)0, c1, false, false);
        }
    }

    const bool keep = ow_n < len1;
#pragma unroll
    for (int r = 0; r < 8; ++r) {
        int row = half16 ? (r + 8) : r;          // C layout: VGPR r -> M=r / r+8
        {
            int o = row;
            float y = c0[r] * sScale[o] + sShift[o];
            y = fminf(fmaxf(y, 0.0f), 20.0f);
            out[(((size_t)b * 32 + o) * 81 + oh) * 512 + ow_n] = keep ? y : 0.0f;
        }
        {
            int o = 16 + row;
            float y = c1[r] * sScale[o] + sShift[o];
            y = fminf(fmaxf(y, 0.0f), 20.0f);
            out[(((size_t)b * 32 + o) * 81 + oh) * 512 + ow_n] = keep ? y : 0.0f;
        }
    }
}

// ---------------------------------------------------------------------------
// Layer 2: Conv2d(32,32,(21,11),s=(2,1),p=(10,5)) + mask/BN/hardtanh fused.
// K = ci(32) x kh x kw(11->12). Weights staged per input channel in LDS.
// ---------------------------------------------------------------------------
__global__ __launch_bounds__(256) void maskcnn_conv2(
    const float* __restrict__ x1,    // [32,32,81,512] (masked layer-1 output)
    const int*   __restrict__ seq,
    const float* __restrict__ w2,    // [32,32,21,11]
    const float* __restrict__ b2,
    const float* __restrict__ g2, const float* __restrict__ be2,
    const float* __restrict__ m2, const float* __restrict__ v2,
    float* __restrict__ out)         // [32,32,41,512]
{
    __shared__ float Wl[32 * 21 * 12];   // one input channel, kw padded to 12
    __shared__ float sScale[32];
    __shared__ float sShift[32];

    const int tid   = threadIdx.x;
    const int b     = blockIdx.z;
    const int oh    = blockIdx.y;
    const int owblk = blockIdx.x * 128;
    const int len1  = ((seq[b] - 1) >> 1) + 1;   // len2 == len1

    if (owblk >= len1) {
        for (int idx = tid; idx < 32 * 128; idx += 256) {
            int o  = idx >> 7;
            int ow = owblk + (idx & 127);
            out[(((size_t)b * 32 + o) * 41 + oh) * 512 + ow] = 0.0f;
        }
        return;
    }

    if (tid < 32) {
        float inv = g2[tid] * __frsqrt_rn(v2[tid] + BN_EPS);
        sScale[tid] = inv;
        sShift[tid] = b2[tid] * inv + be2[tid] - m2[tid] * inv;
    }

    const int wave   = tid >> 5;
    const int lane   = tid & 31;
    const int half16 = lane >> 4;
    const int m      = lane & 15;
    const int ow_n   = owblk + wave * 16 + m;
    const int iwbase = ow_n - 5;

    // ih = 2*oh - 10 + kh in [0,81)  <=>  kh in [10-2*oh, 91-2*oh)
    const int kh_lo = max(0, 10 - 2 * oh);
    const int kh_hi = min(21, 91 - 2 * oh);

    const float* __restrict__ wl0 = Wl + m * 252;
    const float* __restrict__ wl1 = Wl + (m + 16) * 252;

    v8f c0 = {};
    v8f c1 = {};

    for (int ci = 0; ci < 32; ++ci) {
        __syncthreads();                          // protect Wl reuse
        for (int idx = tid; idx < 32 * 252; idx += 256) {
            int o = idx / 252, r = idx - o * 252;
            int kh = r / 12, kw = r - kh * 12;
            Wl[idx] = (kw < 11) ? w2[((size_t)o * 32 + ci) * 231 + kh * 11 + kw]
                                : 0.0f;
        }
        __syncthreads();

        const float* __restrict__ xin = x1 + ((size_t)b * 32 + ci) * 81 * 512;

        for (int kh = kh_lo; kh < kh_hi; ++kh) {
            const int ih = 2 * oh - 10 + kh;      // always in range
            const float* __restrict__ row = xin + ih * 512;
#pragma unroll
            for (int c = 0; c < 3; ++c) {
                const int kb = c * 4 + half16 * 2;
                v2f bf;
#pragma unroll
                for (int j = 0; j < 2; ++j) {
                    int iw = iwbase + kb + j;
                    float v = row[iw & 511];
                    bf[j] = ((unsigned)iw < 512u) ? v : 0.0f;
                }
                v2f a0, a1;
                a0[0] = wl0[kh * 12 + kb]; a0[1] = wl0[kh * 12 + kb + 1];
                a1[0] = wl1[kh * 12 + kb]; a1[1] = wl1[kh * 12 + kb + 1];

                c0 = __builtin_amdgcn_wmma_f32_16x16x4_f32(
                    false, a0, false, bf, (short)0, c0, false, false);
                c1 = __builtin_amdgcn_wmma_f32_16x16x4_f32(
                    false, a1, false, bf, (short)0, c1, false, false);
            }
        }
    }

    const bool keep = ow_n < len1;
#pragma unroll
    for (int r = 0; r < 8; ++r) {
        int row = half16 ? (r + 8) : r;
        {
            int o = row;
            float y = c0[r] * sScale[o] + sShift[o];
            y = fminf(fmaxf(y, 0.0f), 20.0f);
            out[(((size_t)b * 32 + o) * 41 + oh) * 512 + ow_n] = keep ? y : 0.0f;
        }
        {
            int o = 16 + row;
            float y = c1[r] * sScale[o] + sShift[o];
            y = fminf(fmaxf(y, 0.0f), 20.0f);
            out[(((size_t)b * 32 + o) * 41 + oh) * 512 + ow_n] = keep ? y : 0.0f;
        }
    }
}

// Second tuple output: new sequence lengths, stored as raw int32 after the
// float tensor (21,495,808 elements).
__global__ void maskcnn_lens(const int* __restrict__ seq, int* __restrict__ outl)
{
    int b = threadIdx.x;
    if (b < 32) outl[b] = ((seq[b] - 1) >> 1) + 1;
}

extern "C" void kernel_launch(void* const* d_in, const int* in_sizes, int n_in,
                              void* d_out, int out_size, void* d_ws, size_t ws_size,
                              hipStream_t stream) {
    const float* inputs = (const float*)d_in[0];
    const int*   seq    = (const int*)d_in[1];
    const float* w1  = (const float*)d_in[2];
    const float* b1  = (const float*)d_in[3];
    const float* g1  = (const float*)d_in[4];
    const float* be1 = (const float*)d_in[5];
    const float* m1  = (const float*)d_in[6];
    const float* v1  = (const float*)d_in[7];
    const float* w2  = (const float*)d_in[8];
    const float* b2  = (const float*)d_in[9];
    const float* g2  = (const float*)d_in[10];
    const float* be2 = (const float*)d_in[11];
    const float* m2  = (const float*)d_in[12];
    const float* v2  = (const float*)d_in[13];

    float* x1  = (float*)d_ws;                       // 32*32*81*512 floats (~170 MB)
    float* out = (float*)d_out;                      // 32*32*41*512 floats
    int*   outlens = (int*)(out + (size_t)32 * 32 * 41 * 512);

    maskcnn_conv1<<<dim3(4, 81, 32), 256, 0, stream>>>(
        inputs, seq, w1, b1, g1, be1, m1, v1, x1);
    maskcnn_conv2<<<dim3(4, 41, 32), 256, 0, stream>>>(
        x1, seq, w2, b2, g2, be2, m2, v2, out);
    maskcnn_lens<<<1, 32, 0, stream>>>(seq, outlens);
}